// FRN_Projection_31061203484791
// MI455X (gfx1250) — compile-verified
//
#include <hip/hip_runtime.h>
#include <math.h>

typedef float v2f __attribute__((ext_vector_type(2)));
typedef float v8f __attribute__((ext_vector_type(8)));

#define NWAY 5
#define HW 100
#define NS 150
#define NQ 450
#define KROWS 2500        // K_SHOT*PRJ*HW support rows per class
#define IMG_STRIDE 6400   // 64*100 floats per image

// workspace layout (float offsets)
#define WS_STS 0          // 5*64*64
#define WS_GMT 20480      // 5*64*64 (G' transposed: [cls][d][c])
#define WS_NEG 40960      // 450*5

// ---------------------------------------------------------------------------
// Kernel 1: sts[n] = (S^T S)/64 per class via V_WMMA_F32_16X16X4_F32.
// S_raw[k=(s,hw)][c] = feat[(cls*25+s)*6400 + c*100 + hw]; k,k+1 contiguous.
// A tile (c-block = wave*16) has the identical load pattern as a B tile, so
// load it directly -> straight-line loop: 5 global_load_b64 + 4 WMMA.
// ---------------------------------------------------------------------------
__global__ __launch_bounds__(128) void k_sts(const float* __restrict__ feat,
                                             float* __restrict__ sts) {
  const int cls   = blockIdx.x;
  const int tid   = threadIdx.x;
  const int lane  = tid & 31;
  const int wave  = tid >> 5;            // 4 waves -> c-block = wave*16
  const int l15   = lane & 15;
  const int kpair = (lane >> 4) << 1;    // 0 or 2 (K position of this half-wave)
  const int aoff  = (wave * 16 + l15) * 100;

  const float* base = feat + (size_t)cls * 25 * IMG_STRIDE;

  v8f acc0 = {}, acc1 = {}, acc2 = {}, acc3 = {};

  for (int k0 = 0; k0 < KROWS; k0 += 4) {
    int kp = k0 + kpair;
    int s  = kp / 100;
    int hw = kp - s * 100;               // hw, hw+1 stay inside one image (k0 % 4 == 0)
    const float* p = base + (size_t)s * IMG_STRIDE + hw;
    v2f a  = *(const v2f*)(p + aoff);
    v2f t0 = *(const v2f*)(p + (0  + l15) * 100);
    v2f t1 = *(const v2f*)(p + (16 + l15) * 100);
    v2f t2 = *(const v2f*)(p + (32 + l15) * 100);
    v2f t3 = *(const v2f*)(p + (48 + l15) * 100);
    acc0 = __builtin_amdgcn_wmma_f32_16x16x4_f32(false, a, false, t0, (short)0, acc0, false, false);
    acc1 = __builtin_amdgcn_wmma_f32_16x16x4_f32(false, a, false, t1, (short)0, acc1, false, false);
    acc2 = __builtin_amdgcn_wmma_f32_16x16x4_f32(false, a, false, t2, (short)0, acc2, false, false);
    acc3 = __builtin_amdgcn_wmma_f32_16x16x4_f32(false, a, false, t3, (short)0, acc3, false, false);
  }

  // C/D layout: VGPR v, lanes 0-15 -> M=v, lanes 16-31 -> M=v+8; N = lane&15
  const int c0 = wave * 16;
  const int mh = (lane >> 4) * 8;
  float* o = sts + (size_t)cls * 4096;
  const float sc = 1.0f / 64.0f;         // fold both 1/sqrt(64) scales
#pragma unroll
  for (int v = 0; v < 8; ++v) {
    int row = (c0 + v + mh) * 64;
    o[row + 0  + l15] = acc0[v] * sc;
    o[row + 16 + l15] = acc1[v] * sc;
    o[row + 32 + l15] = acc2[v] * sc;
    o[row + 48 + l15] = acc3[v] * sc;
  }
}

// ---------------------------------------------------------------------------
// Kernel 2: Gauss-Jordan on [sts+lam*I | sts]  ->  hat = (sts+lam*I)^-1 sts.
// Emits G'^T[d][c] = (rho*hat[c][d] - delta(c,d)) / 8  (query scale folded in).
// ---------------------------------------------------------------------------
__global__ __launch_bounds__(128) void k_inv(const float* __restrict__ sts,
                                             const float* __restrict__ r,
                                             float* __restrict__ gmt) {
  __shared__ float aug[64][128];
  __shared__ float colp[64];
  const int cls = blockIdx.x;
  const int tid = threadIdx.x;

  const float lam = (2500.0f / 64.0f) * __expf(r[0]);
  const float rho = __expf(r[1]);

  const float* S = sts + (size_t)cls * 4096;
  for (int i = tid; i < 64 * 128; i += 128) {
    int row = i >> 7, col = i & 127;
    float v = S[row * 64 + (col & 63)];
    if (col < 64 && col == row) v += lam;
    aug[row][col] = v;
  }
  __syncthreads();

  for (int p = 0; p < 64; ++p) {
    if (tid < 64) colp[tid] = aug[tid][p];   // snapshot pivot column
    __syncthreads();
    {
      float pr = aug[p][tid] / colp[p];      // each thread owns column `tid`
      aug[p][tid] = pr;
#pragma unroll 4
      for (int row = 0; row < 64; ++row)
        if (row != p) aug[row][tid] -= colp[row] * pr;
    }
    __syncthreads();
  }

  float* O = gmt + (size_t)cls * 4096;
  for (int i = tid; i < 4096; i += 128) {
    int c = i >> 6, d = i & 63;
    float h = aug[c][64 + d];                // hat[c][d]
    O[d * 64 + c] = (rho * h - (c == d ? 1.0f : 0.0f)) * 0.125f;
  }
}

// ---------------------------------------------------------------------------
// Kernel 3: per query image, P = Q_raw * G' (WMMA f32 16x16x4, K=64), fused
// squared-row-norm epilogue, mean over 100 hw -> neg_l2_dist[img][cls].
// 224 threads = 7 waves, one 16-row hw tile each (rows 100..111 zero-padded);
// no masked-EXEC region around the WMMAs. G' staged in LDS (80 KB of 320 KB).
// Deterministic reductions only.
// ---------------------------------------------------------------------------
__global__ __launch_bounds__(224) void k_dist(const float* __restrict__ feat,
                                              const float* __restrict__ gmt,
                                              float* __restrict__ neg) {
  __shared__ float gs[NWAY * 4096];      // 80 KB
  __shared__ float wsum[7][NWAY];
  const int img   = blockIdx.x;
  const int tid   = threadIdx.x;
  const int lane  = tid & 31;
  const int wave  = tid >> 5;            // 0..6, one hw tile per wave
  const int l15   = lane & 15;
  const int kpair = (lane >> 4) << 1;

  for (int i = tid; i < NWAY * 4096; i += 224) gs[i] = gmt[i];
  __syncthreads();

  const int  hw = wave * 16 + l15;
  const bool ok = hw < HW;               // zero-pad rows 100..111 -> contribute 0
  const float* qb = feat + (size_t)(NS + img) * IMG_STRIDE + hw;

  // cache all A k-tiles: a[k].{x,y} = Q_raw[hw][4k+kpair(+1)]
  v2f a[16];
#pragma unroll
  for (int k = 0; k < 16; ++k) {
    int kc = k * 4 + kpair;
    a[k].x = ok ? qb[(kc + 0) * 100] : 0.0f;
    a[k].y = ok ? qb[(kc + 1) * 100] : 0.0f;
  }

#pragma unroll 1
  for (int cls = 0; cls < NWAY; ++cls) {
    const float* g = gs + cls * 4096;
    v8f sq = {};
#pragma unroll 1
    for (int dt = 0; dt < 4; ++dt) {
      const float* gb = g + (dt * 16 + l15) * 64 + kpair;  // G'^T[d][kc], contiguous in kc
      v8f c = {};
#pragma unroll
      for (int k = 0; k < 16; ++k) {
        v2f b = *(const v2f*)(gb + k * 4);
        c = __builtin_amdgcn_wmma_f32_16x16x4_f32(false, a[k], false, b, (short)0, c, false, false);
      }
      sq += c * c;                       // accumulate P^2 across d-tiles
    }
    // sum over the 16 lanes of each half (columns), then over the 8 rows/half,
    // then combine halves -> full 16x64 tile sum (fixed order = deterministic)
    float tot = 0.0f;
#pragma unroll
    for (int v = 0; v < 8; ++v) {
      float x = sq[v];
      x += __shfl_xor(x, 1, 32);
      x += __shfl_xor(x, 2, 32);
      x += __shfl_xor(x, 4, 32);
      x += __shfl_xor(x, 8, 32);
      tot += x;
    }
    tot += __shfl_xor(tot, 16, 32);
    if (lane == 0) wsum[wave][cls] = tot;
  }

  __syncthreads();
  if (tid < NWAY) {
    float s = 0.0f;
#pragma unroll
    for (int w = 0; w < 7; ++w) s += wsum[w][tid];
    neg[img * NWAY + tid] = -s * (1.0f / (float)HW);
  }
}

// ---------------------------------------------------------------------------
// Kernel 4: y_pred = softmax(neg); loss = NLL(log_softmax(neg*scale), label).
// ---------------------------------------------------------------------------
__global__ __launch_bounds__(512) void k_softmax(const float* __restrict__ neg,
                                                 const int* __restrict__ label,
                                                 const float* __restrict__ scale,
                                                 float* __restrict__ out) {
  __shared__ float lsum[512];
  const int tid = threadIdx.x;
  float contrib = 0.0f;
  if (tid < NQ) {
    float v[NWAY];
#pragma unroll
    for (int n = 0; n < NWAY; ++n) v[n] = neg[tid * NWAY + n];

    float mx = v[0];
#pragma unroll
    for (int n = 1; n < NWAY; ++n) mx = fmaxf(mx, v[n]);
    float e[NWAY], se = 0.0f;
#pragma unroll
    for (int n = 0; n < NWAY; ++n) { e[n] = __expf(v[n] - mx); se += e[n]; }
    float inv = 1.0f / se;
#pragma unroll
    for (int n = 0; n < NWAY; ++n) out[tid * NWAY + n] = e[n] * inv;

    float sc = scale[0];
    float ml = v[0] * sc;
#pragma unroll
    for (int n = 1; n < NWAY; ++n) ml = fmaxf(ml, v[n] * sc);
    float sl = 0.0f;
#pragma unroll
    for (int n = 0; n < NWAY; ++n) sl += __expf(v[n] * sc - ml);
    float lse = __logf(sl) + ml;
    int lb = label[tid];
    contrib = -(v[lb] * sc - lse);
  }
  lsum[tid] = contrib;
  __syncthreads();
  for (int s = 256; s > 0; s >>= 1) {
    if (tid < s) lsum[tid] += lsum[tid + s];
    __syncthreads();
  }
  if (tid == 0) out[NQ * NWAY] = lsum[0] * (1.0f / (float)NQ);
}

// ---------------------------------------------------------------------------
extern "C" void kernel_launch(void* const* d_in, const int* in_sizes, int n_in,
                              void* d_out, int out_size, void* d_ws, size_t ws_size,
                              hipStream_t stream) {
  const float* feat  = (const float*)d_in[0];
  const int*   label = (const int*)d_in[1];
  const float* scale = (const float*)d_in[2];
  const float* r     = (const float*)d_in[3];

  float* ws  = (float*)d_ws;
  float* sts = ws + WS_STS;
  float* gmt = ws + WS_GMT;
  float* neg = ws + WS_NEG;
  float* out = (float*)d_out;

  k_sts    <<<NWAY, 128, 0, stream>>>(feat, sts);
  k_inv    <<<NWAY, 128, 0, stream>>>(sts, r, gmt);
  k_dist   <<<NQ,   224, 0, stream>>>(feat, gmt, neg);
  k_softmax<<<1,    512, 0, stream>>>(neg, label, scale, out);
}